// EncP_38585986187437
// MI455X (gfx1250) — compile-verified
//
#include <hip/hip_runtime.h>
#include <math.h>
#include <float.h>

#define B_    16
#define N_    2048
#define KNB   40
#define EMBED 72

typedef float v2f __attribute__((ext_vector_type(2)));
typedef float v8f __attribute__((ext_vector_type(8)));

// ---------------------------------------------------------------------------
// Embed: e[b,o,n] = sum_c x[b,c,n] * ew[o,c]   (C=3, scalar is fine)
// ---------------------------------------------------------------------------
__global__ void k_embed(const float* __restrict__ x, const float* __restrict__ ew,
                        float* __restrict__ e) {
  int id = blockIdx.x * blockDim.x + threadIdx.x;
  int total = B_ * EMBED * N_;
  if (id >= total) return;
  int n = id % N_;
  int o = (id / N_) % EMBED;
  int b = id / (N_ * EMBED);
  const float* xb = x + (size_t)b * 3 * N_;
  e[id] = xb[0 * N_ + n] * ew[o * 3 + 0]
        + xb[1 * N_ + n] * ew[o * 3 + 1]
        + xb[2 * N_ + n] * ew[o * 3 + 2];
}

// ---------------------------------------------------------------------------
// Per-channel BN statistics over (batch, spatial): x is [B, C, P]
// ---------------------------------------------------------------------------
__global__ __launch_bounds__(256) void k_bn_stats(const float* __restrict__ x,
                                                  int C, int P,
                                                  float* __restrict__ mean,
                                                  float* __restrict__ rstd) {
  int c = blockIdx.x;
  int t = threadIdx.x;
  __shared__ float ssum[256], ssq[256];
  float s = 0.f, q = 0.f;
  for (int b = 0; b < B_; ++b) {
    const float* p = x + ((size_t)b * C + c) * P;
    for (int i = t; i < P; i += 256) { float v = p[i]; s += v; q += v * v; }
  }
  ssum[t] = s; ssq[t] = q; __syncthreads();
  for (int st = 128; st > 0; st >>= 1) {
    if (t < st) { ssum[t] += ssum[t + st]; ssq[t] += ssq[t + st]; }
    __syncthreads();
  }
  if (t == 0) {
    float cnt = (float)B_ * (float)P;
    float m = ssum[0] / cnt;
    float v = ssq[0] / cnt - m * m;          // biased variance (jnp.var)
    mean[c] = m;
    rstd[c] = rsqrtf(v + 1e-5f);
  }
}

// BN apply (+optional residual) + ReLU, in place on x [B,C,P]
__global__ void k_bn_relu(float* __restrict__ x, const float* __restrict__ mean,
                          const float* __restrict__ rstd, const float* __restrict__ gm,
                          const float* __restrict__ bt, const float* __restrict__ res,
                          int C, int P) {
  size_t id = (size_t)blockIdx.x * blockDim.x + threadIdx.x;
  size_t total = (size_t)B_ * C * P;
  if (id >= total) return;
  int c = (int)((id / (size_t)P) % C);
  float v = (x[id] - mean[c]) * rstd[c] * gm[c] + bt[c];
  if (res) v += res[id];
  x[id] = fmaxf(v, 0.f);
}

// ---------------------------------------------------------------------------
// Deterministic "random" sampling + gather of lc_xyz
// ---------------------------------------------------------------------------
__global__ void k_sample(const float* __restrict__ xyz, int n_cur, int g,
                         int* __restrict__ sidx, float* __restrict__ lc_xyz,
                         unsigned seed) {
  int id = blockIdx.x * blockDim.x + threadIdx.x;
  if (id >= B_ * g) return;
  int gi = id % g, b = id / g;
  unsigned h = seed ^ (unsigned)(b * 2654435761u) ^ (unsigned)(gi * 40503u + 0x9E3779B9u);
  h ^= h >> 16; h *= 0x7feb352du; h ^= h >> 15; h *= 0x846ca68bu; h ^= h >> 16;
  int idx = (int)(h % (unsigned)n_cur);
  sidx[id] = idx;
  const float* p = xyz + ((size_t)b * n_cur + idx) * 3;
  float* q = lc_xyz + (size_t)id * 3;
  q[0] = p[0]; q[1] = p[1]; q[2] = p[2];
}

// ---------------------------------------------------------------------------
// kNN: one block per (b,g); d^2 table in LDS, K rounds of arg-min reduction
// ---------------------------------------------------------------------------
__global__ __launch_bounds__(256) void k_knn(const float* __restrict__ xyz,
                                             const float* __restrict__ lc_xyz,
                                             int n_cur, int g,
                                             int* __restrict__ knn_idx) {
  __shared__ float sd[N_];
  __shared__ float rmin[256];
  __shared__ int   rarg[256];
  int bg = blockIdx.x;
  int b  = bg / g;
  int t  = threadIdx.x;
  const float* q = lc_xyz + (size_t)bg * 3;
  float qx = q[0], qy = q[1], qz = q[2];
  const float* xb = xyz + (size_t)b * n_cur * 3;
  for (int j = t; j < n_cur; j += 256) {
    float dx = xb[j * 3 + 0] - qx, dy = xb[j * 3 + 1] - qy, dz = xb[j * 3 + 2] - qz;
    sd[j] = dx * dx + dy * dy + dz * dz;
  }
  __syncthreads();
  int* out = knn_idx + (size_t)bg * KNB;
  for (int k = 0; k < KNB; ++k) {
    float lm = FLT_MAX; int la = 0;
    for (int j = t; j < n_cur; j += 256) {
      float v = sd[j];
      if (v < lm) { lm = v; la = j; }
    }
    rmin[t] = lm; rarg[t] = la; __syncthreads();
    for (int s = 128; s > 0; s >>= 1) {
      if (t < s && rmin[t + s] < rmin[t]) { rmin[t] = rmin[t + s]; rarg[t] = rarg[t + s]; }
      __syncthreads();
    }
    if (t == 0) { out[k] = rarg[0]; sd[rarg[0]] = FLT_MAX; }
    __syncthreads();
  }
}

// ---------------------------------------------------------------------------
// Global std of diff (ddof=1): accumulate sum / sumsq
// ---------------------------------------------------------------------------
__global__ void k_zero2(float* s) { if (threadIdx.x < 2) s[threadIdx.x] = 0.f; }

__global__ __launch_bounds__(256) void k_diff_stats(const float* __restrict__ xyz,
                                                    const float* __restrict__ lc_xyz,
                                                    const int* __restrict__ knn_idx,
                                                    int n_cur, int g,
                                                    float* __restrict__ stats) {
  int id = blockIdx.x * blockDim.x + threadIdx.x;
  int total = B_ * g * KNB;
  float s = 0.f, q = 0.f;
  if (id < total) {
    int k = id % KNB; int bg = id / KNB;
    int b = bg / g;
    int j = knn_idx[(size_t)bg * KNB + k];
    const float* p = xyz + ((size_t)b * n_cur + j) * 3;
    const float* c = lc_xyz + (size_t)bg * 3;
    for (int a = 0; a < 3; ++a) { float d = p[a] - c[a]; s += d; q += d * d; }
  }
  __shared__ float ss[256], sq[256];
  int t = threadIdx.x;
  ss[t] = s; sq[t] = q; __syncthreads();
  for (int st = 128; st > 0; st >>= 1) {
    if (t < st) { ss[t] += ss[t + st]; sq[t] += sq[t + st]; }
    __syncthreads();
  }
  if (t == 0) { atomicAdd(&stats[0], ss[0]); atomicAdd(&stats[1], sq[0]); }
}

// ---------------------------------------------------------------------------
// Build w = concat(knn_x, lc_x) + PosE_Geo     w: [B, Dout, g, K]
// ---------------------------------------------------------------------------
__global__ void k_build_w(const float* __restrict__ xyz, const float* __restrict__ lc_xyz,
                          const int* __restrict__ sidx, const int* __restrict__ knn_idx,
                          const float* __restrict__ feats, const float* __restrict__ stats,
                          int n_cur, int g, int Cin, int Dout,
                          float* __restrict__ w) {
  size_t id = (size_t)blockIdx.x * blockDim.x + threadIdx.x;
  size_t total = (size_t)B_ * Dout * g * KNB;
  if (id >= total) return;
  int k  = (int)(id % KNB);
  int gi = (int)((id / KNB) % g);
  int ch = (int)((id / ((size_t)KNB * g)) % Dout);
  int b  = (int)(id / ((size_t)KNB * g * Dout));
  size_t bg = (size_t)b * g + gi;
  int j = knn_idx[bg * KNB + k];
  // feature part: channels [0,Cin) = neighbor feats, [Cin,2Cin) = center feats
  float fv;
  if (ch < Cin) fv = feats[((size_t)b * Cin + ch) * n_cur + j];
  else          fv = feats[((size_t)b * Cin + (ch - Cin)) * n_cur + sidx[bg]];
  // positional encoding
  float cnt = (float)B_ * g * KNB * 3.0f;
  float sm = stats[0], sq = stats[1];
  float var = (sq - sm * sm / cnt) / (cnt - 1.0f);   // ddof=1
  float stdv = sqrtf(fmaxf(var, 0.f)) + 1e-5f;
  int f = Dout / 6;
  int twof = 2 * f;
  int a  = ch / twof;          // coord 0..2
  int tt = ch % twof;
  int tp = (tt < f) ? tt : tt - f;
  float dcoord = (xyz[((size_t)b * n_cur + j) * 3 + a] - lc_xyz[bg * 3 + a]) / stdv;
  float dim_embed = __powf(1000.0f, (float)tp / (float)f);
  float dv = 100.0f * dcoord / dim_embed;
  float pe = (tt < f) ? __sinf(dv) : __cosf(dv);
  w[id] = fv + pe;
}

// ---------------------------------------------------------------------------
// fp32 WMMA GEMM:  Y[b] = W (OxC) * X[b] (CxP) + bias
//   Block = 8 waves. M tile = 16 rows (A strip staged in LDS, zero-padded),
//   N tile = 512 cols: wave w -> 64 cols = four 16x16 accumulators, so one
//   ds_load_2addr A-fragment feeds 4 WMMAs. B addressed as uniform SGPR base
//   + 32-bit lane offset (GVS mode) to kill 64-bit VALU address math.
//   Requires: C % 4 == 0, P % 512 == 0 (holds for all stages).
//   Dynamic LDS = C*16 floats.
// ---------------------------------------------------------------------------
__global__ __launch_bounds__(256) void k_gemm_wmma(const float* __restrict__ W,
                                                   const float* __restrict__ X,
                                                   const float* __restrict__ bias,
                                                   float* __restrict__ Y,
                                                   int O, int C, int P) {
  extern __shared__ float As[];            // [C][16]  (As[c*16 + m])
  const int b   = blockIdx.z;
  const int m0  = blockIdx.x * 16;
  const int p0  = blockIdx.y * 512;
  const int tid = threadIdx.x;

  // cooperative fill of the 16xC weight strip (coalesced W reads),
  // transposed into As[c*16+m], zero-padded for ragged O
  for (int idx = tid; idx < 16 * C; idx += 256) {
    int m = idx / C, c = idx - m * C;
    int gm = m0 + m;
    As[c * 16 + m] = (gm < O) ? W[(size_t)gm * C + c] : 0.0f;
  }
  __syncthreads();

  const int wave = tid >> 5;
  const int lane = tid & 31;
  const int half = lane >> 4;              // 0: K={0,1}, 1: K={2,3}
  const int l    = lane & 15;

  const float* Xb = X + (size_t)b * C * P; // uniform per block -> SGPR base
  const int n    = p0 + wave * 64 + l;     // first column of this lane
  int off0 = half * 2 * P + n;             // row (c0+2*half), col n  (32-bit)
  int off1 = off0 + P;                     // next K row
  const float* Ap = As + half * 32 + l;    // As[(half*2)*16 + l]

  v8f acc0 = {}, acc1 = {}, acc2 = {}, acc3 = {};
  for (int c0 = 0; c0 < C; c0 += 4) {
    v2f a;
    a.x = Ap[0];                           // A[m][c0+2*half]
    a.y = Ap[16];                          // A[m][c0+2*half+1]
    v2f b0, b1, b2, b3;
    b0.x = Xb[off0];      b0.y = Xb[off1];
    b1.x = Xb[off0 + 16]; b1.y = Xb[off1 + 16];
    b2.x = Xb[off0 + 32]; b2.y = Xb[off1 + 32];
    b3.x = Xb[off0 + 48]; b3.y = Xb[off1 + 48];
    acc0 = __builtin_amdgcn_wmma_f32_16x16x4_f32(false, a, false, b0,
                                                 (short)0, acc0, false, false);
    acc1 = __builtin_amdgcn_wmma_f32_16x16x4_f32(false, a, false, b1,
                                                 (short)0, acc1, false, false);
    acc2 = __builtin_amdgcn_wmma_f32_16x16x4_f32(false, a, false, b2,
                                                 (short)0, acc2, false, false);
    acc3 = __builtin_amdgcn_wmma_f32_16x16x4_f32(false, a, false, b3,
                                                 (short)0, acc3, false, false);
    Ap   += 64;                            // 4 K-rows * 16
    off0 += 4 * P;
    off1 += 4 * P;
  }

  float* Yb = Y + (size_t)b * O * P;       // uniform -> SGPR base
#pragma unroll
  for (int r = 0; r < 8; ++r) {
    int m = m0 + r + half * 8;             // C/D layout: VGPR r -> rows r, r+8
    if (m < O) {
      float bv = bias[m];
      int yoff = m * P + n;                // 32-bit offset, GVS form
      Yb[yoff]      = acc0[r] + bv;
      Yb[yoff + 16] = acc1[r] + bv;
      Yb[yoff + 32] = acc2[r] + bv;
      Yb[yoff + 48] = acc3[r] + bv;
    }
  }
}

// ---------------------------------------------------------------------------
// Max-pool over K: feats_out[b,o,g] = max_k w[b,o,g,k]
// ---------------------------------------------------------------------------
__global__ void k_maxpool(const float* __restrict__ w, int Dout, int g,
                          float* __restrict__ feats_out) {
  int id = blockIdx.x * blockDim.x + threadIdx.x;
  int total = B_ * Dout * g;
  if (id >= total) return;
  const float* p = w + (size_t)id * KNB;
  float m = p[0];
  for (int k = 1; k < KNB; ++k) m = fmaxf(m, p[k]);
  feats_out[id] = m;
}

__global__ void k_copy_out(const float* __restrict__ xyz, int g,
                           const float* __restrict__ feats, int Dout,
                           float* __restrict__ out) {
  int id = blockIdx.x * blockDim.x + threadIdx.x;
  int nxyz = B_ * g * 3;
  int nf = B_ * Dout * g;
  if (id < nxyz) out[id] = xyz[id];
  else if (id < nxyz + nf) out[id] = feats[id - nxyz];
}

// ---------------------------------------------------------------------------
extern "C" void kernel_launch(void* const* d_in, const int* in_sizes, int n_in,
                              void* d_out, int out_size, void* d_ws, size_t ws_size,
                              hipStream_t stream) {
  (void)in_sizes; (void)n_in; (void)out_size; (void)ws_size;
  const float* xyz_in  = (const float*)d_in[0];   // [16,2048,3]
  const float* x_in    = (const float*)d_in[1];   // [16,3,2048]
  const float* embed_w = (const float*)d_in[2];   // [72,3]
  const float* embed_g = (const float*)d_in[3];
  const float* embed_b = (const float*)d_in[4];

  // ---- workspace carve (bytes, 256-aligned) ----
  char* p = (char*)d_ws;
  auto alloc = [&](size_t bytes) -> void* {
    void* r = (void*)p;
    p += (bytes + 255) & ~(size_t)255;
    return r;
  };
  float* wbuf  = (float*)alloc((size_t)94371840 * 4);  // B*144*1024*40 (constant/stage)
  float* y2buf = (float*)alloc((size_t)94371840 * 4);
  float* hbuf  = (float*)alloc((size_t)47185920 * 4);
  float* fA    = (float*)alloc((size_t)2359296 * 4);
  float* fB    = (float*)alloc((size_t)2359296 * 4);
  float* xA    = (float*)alloc((size_t)49152 * 4);
  float* xB    = (float*)alloc((size_t)49152 * 4);
  int*   sidx  = (int*)alloc((size_t)16384 * 4);
  int*   knn   = (int*)alloc((size_t)655360 * 4);
  float* mean  = (float*)alloc((size_t)1152 * 4);
  float* rstd  = (float*)alloc((size_t)1152 * 4);
  float* stats = (float*)alloc(256);

  // ---- embed ----
  {
    int tot = B_ * EMBED * N_;
    k_embed<<<(tot + 255) / 256, 256, 0, stream>>>(x_in, embed_w, fA);
    k_bn_stats<<<EMBED, 256, 0, stream>>>(fA, EMBED, N_, mean, rstd);
    k_bn_relu<<<(tot + 255) / 256, 256, 0, stream>>>(fA, mean, rstd, embed_g,
                                                     embed_b, nullptr, EMBED, N_);
  }

  const int OD[4] = {144, 288, 576, 1152};
  const int HD[4] = {72, 144, 32, 576};

  const float* xyz_cur = xyz_in;
  float* xyz_buf[2] = {xA, xB};
  float* feats_cur = fA;
  float* feats_next = fB;
  int n_cur = N_;
  int Cin = EMBED;

  for (int i = 0; i < 4; ++i) {
    int g = n_cur / 2, O = OD[i], H = HD[i], P = g * KNB;
    const float* w1  = (const float*)d_in[5 + i * 8 + 0];  // [H,O]
    const float* b1  = (const float*)d_in[5 + i * 8 + 1];
    const float* g1  = (const float*)d_in[5 + i * 8 + 2];
    const float* bb1 = (const float*)d_in[5 + i * 8 + 3];
    const float* w2  = (const float*)d_in[5 + i * 8 + 4];  // [O,H]
    const float* b2  = (const float*)d_in[5 + i * 8 + 5];
    const float* g2  = (const float*)d_in[5 + i * 8 + 6];
    const float* bb2 = (const float*)d_in[5 + i * 8 + 7];

    float* lc = xyz_buf[i & 1];
    k_sample<<<(B_ * g + 255) / 256, 256, 0, stream>>>(xyz_cur, n_cur, g, sidx, lc,
                                                       1000u + (unsigned)i);
    k_knn<<<B_ * g, 256, 0, stream>>>(xyz_cur, lc, n_cur, g, knn);
    k_zero2<<<1, 32, 0, stream>>>(stats);
    int tot_k = B_ * g * KNB;
    k_diff_stats<<<(tot_k + 255) / 256, 256, 0, stream>>>(xyz_cur, lc, knn, n_cur, g, stats);

    size_t tot_w = (size_t)B_ * O * P;
    k_build_w<<<(unsigned)((tot_w + 255) / 256), 256, 0, stream>>>(
        xyz_cur, lc, sidx, knn, feats_cur, stats, n_cur, g, Cin, O, wbuf);

    // ---- Linear2Layer: GEMM1 -> BN -> ReLU -> GEMM2 -> BN -> +res -> ReLU ----
    dim3 grid1((H + 15) / 16, P / 512, B_);
    k_gemm_wmma<<<grid1, 256, (size_t)O * 16 * 4, stream>>>(w1, wbuf, b1, hbuf, H, O, P);
    k_bn_stats<<<H, 256, 0, stream>>>(hbuf, H, P, mean, rstd);
    size_t tot_h = (size_t)B_ * H * P;
    k_bn_relu<<<(unsigned)((tot_h + 255) / 256), 256, 0, stream>>>(
        hbuf, mean, rstd, g1, bb1, nullptr, H, P);

    dim3 grid2((O + 15) / 16, P / 512, B_);
    k_gemm_wmma<<<grid2, 256, (size_t)H * 16 * 4, stream>>>(w2, hbuf, b2, y2buf, O, H, P);
    k_bn_stats<<<O, 256, 0, stream>>>(y2buf, O, P, mean, rstd);
    k_bn_relu<<<(unsigned)((tot_w + 255) / 256), 256, 0, stream>>>(
        y2buf, mean, rstd, g2, bb2, wbuf, O, P);

    k_maxpool<<<(unsigned)((B_ * O * g + 255) / 256), 256, 0, stream>>>(
        y2buf, O, g, feats_next);

    xyz_cur = lc;
    n_cur = g;
    Cin = O;
    float* t = feats_cur; feats_cur = feats_next; feats_next = t;
  }

  // ---- output: (xyz [16,128,3], feats [16,1152,128]) concatenated ----
  int tot_out = B_ * 128 * 3 + B_ * 1152 * 128;
  k_copy_out<<<(tot_out + 255) / 256, 256, 0, stream>>>(xyz_cur, 128, feats_cur,
                                                        1152, (float*)d_out);
}